// EKFPositionLayer_14783277433316
// MI455X (gfx1250) — compile-verified
//
#include <hip/hip_runtime.h>

typedef float v2f __attribute__((ext_vector_type(2)));
typedef float v8f __attribute__((ext_vector_type(8)));

#define TLEN 1024
#define NBATCH 8192
#define DTC (1.0f/30.0f)
#define GRAV 9.81f

// ---------------------------------------------------------------------------
// Kernel A: lane-parallel Riccati recursion. 16 lanes each own element (i,j)
// of the 4x4 matrices; produces K_t (row-major 16 floats) for t=0..1023.
// ---------------------------------------------------------------------------
__global__ void __launch_bounds__(32) ekf_gains(float* __restrict__ ktab)
{
    const int lane = threadIdx.x & 31;
    const int e = lane & 15;
    const int i = e >> 2, j = e & 3;
    const float dt = DTC;
    const float Qd = (i == j) ? ((i < 2) ? 0.01f : 0.1f) : 0.0f;
    const float Rd = (i == j) ? ((i < 2) ? 5.0f  : 1.0f) : 0.0f;
    float P = (i == j) ? 1.0f : 0.0f;

    for (int t = 0; t < TLEN; ++t) {
        // P = F P F^T + Q  (F = I + dt*E, E[0][2]=E[1][3]=1)
        float pr = __shfl(P, (j < 2) ? (e + 2) : e, 32);
        float A1 = P + ((j < 2) ? dt * pr : 0.0f);
        float ar = __shfl(A1, (i < 2) ? (e + 8) : e, 32);
        P = A1 + ((i < 2) ? dt * ar : 0.0f) + Qd;

        // S = P + R ; M = S^{-1} via Gauss-Jordan (S is SPD, no pivoting)
        float S = P + Rd;
        float M = (i == j) ? 1.0f : 0.0f;
        #pragma unroll
        for (int k = 0; k < 4; ++k) {
            float piv = __shfl(S, k * 4 + k, 32);
            float rs  = 1.0f / piv;
            float skj = __shfl(S, k * 4 + j, 32) * rs;
            float mkj = __shfl(M, k * 4 + j, 32) * rs;
            float f   = __shfl(S, i * 4 + k, 32);
            if (i == k) { S = skj;          M = mkj; }
            else        { S = S - f * skj;  M = M - f * mkj; }
        }

        // K = (S^{-1} P)^T : K[i][j] = sum_k M[j][k] * P[k][i]
        float K = 0.0f;
        #pragma unroll
        for (int k = 0; k < 4; ++k)
            K += __shfl(M, j * 4 + k, 32) * __shfl(P, k * 4 + i, 32);

        // P = (I - K) P
        float KP = 0.0f;
        #pragma unroll
        for (int k = 0; k < 4; ++k)
            KP += __shfl(K, i * 4 + k, 32) * __shfl(P, k * 4 + j, 32);
        P = P - KP;

        if (lane < 16) ktab[t * 16 + e] = K;
    }
}

// ---------------------------------------------------------------------------
// Kernel B: streaming filter. One wave = 16 batches; per step one
// v_wmma_f32_16x16x4_f32 applies   D = K_t * innov + x_pred.
//   A (16x4): rows 0..3 = K_t, rows 4..15 = 0
//   B (4x16): column n = innovation of batch n
//             (lane n: innov[0..1], lane n+16: innov[2..3])
//   C/D (16x16): lane n VGPR r = state comp r of batch n (rows 4..15 = 0)
// Gains are broadcast into LDS via async load-to-LDS.
// ---------------------------------------------------------------------------
__global__ void __launch_bounds__(128) ekf_filter(const float* __restrict__ pos,
                                                  const float* __restrict__ ktab,
                                                  float* __restrict__ out)
{
    __shared__ float sK[TLEN * 16];          // 64 KB gain table

    const unsigned tid = threadIdx.x;
    const unsigned lds_base = (unsigned)(size_t)(&sK[0]);
    // async copy 64 KB: 4096 x b128, 32 per thread
    for (int jj = 0; jj < 32; ++jj) {
        unsigned idx4 = (unsigned)jj * 128u + tid;
        unsigned lds_addr = lds_base + idx4 * 16u;
        unsigned long long gaddr = (unsigned long long)(ktab + (size_t)idx4 * 4u);
        asm volatile("global_load_async_to_lds_b128 %0, %1, off"
                     :: "v"(lds_addr), "v"(gaddr) : "memory");
    }
    asm volatile("s_wait_asynccnt 0" ::: "memory");
    __syncthreads();

    const int lane  = tid & 31;
    const int wave  = tid >> 5;
    const int n     = lane & 15;
    const bool lower = lane < 16;
    const int batch = blockIdx.x * 64 + wave * 16 + n;

    const float dt = DTC, inv_dt = 30.0f;
    const float u1 = 0.5f * GRAV * DTC * DTC;   // gravity term on py
    const float u3 = GRAV * DTC;                // gravity term on vy

    const float* __restrict__ pin  = pos + (size_t)batch * (TLEN * 3);
    float*       __restrict__ pout = out + (size_t)batch * (TLEN * 3);

    v8f c = {0.f, 0.f, 0.f, 0.f, 0.f, 0.f, 0.f, 0.f};
    float pmx = 0.f, pmy = 0.f;

    for (int t = 0; t < TLEN; ++t) {
        const float* p = pin + t * 3;
        float mx = p[0], my = p[1];
        float mz = 0.f;
        if (lower) mz = p[2];

        // predict: x_pred = F x + u (meaningful on lower half)
        float x0 = c[0], x1 = c[1], x2 = c[2], x3 = c[3];
        float px0 = x0 + dt * x2;
        float px1 = x1 + dt * x3 + u1;
        float px2 = x2;
        float px3 = x3 + u3;

        // ship predicted velocity comps to the upper half (lane n -> n+16)
        float s2 = __shfl(px2, n, 32);
        float s3 = __shfl(px3, n, 32);

        // velocity measurement (v_0 = 0)
        float vmx = (t == 0) ? 0.f : (mx - pmx) * inv_dt;
        float vmy = (t == 0) ? 0.f : (my - pmy) * inv_dt;
        pmx = mx; pmy = my;

        // B operand: innovation column per batch
        float b0 = lower ? (mx - px0) : (vmx - s2);
        float b1 = lower ? (my - px1) : (vmy - s3);

        // A operand: gain row (lower: K[n][0..1], upper: K[n][2..3]); 0 for n>=4
        int r = (n < 4) ? n : 3;
        const float* kp = &sK[t * 16 + r * 4 + (lower ? 0 : 2)];
        float a0 = kp[0], a1 = kp[1];
        if (n >= 4) { a0 = 0.f; a1 = 0.f; }

        v2f a = {a0, a1};
        v2f b = {b0, b1};
        c[0] = lower ? px0 : 0.f;
        c[1] = lower ? px1 : 0.f;
        c[2] = lower ? px2 : 0.f;
        c[3] = lower ? px3 : 0.f;

        // D = A*B + C  ->  x_new = x_pred + K * innov (rows 0..3)
        c = __builtin_amdgcn_wmma_f32_16x16x4_f32(false, a, false, b,
                                                  (short)0, c, false, false);

        if (lower) {
            float* o = pout + t * 3;
            o[0] = c[0];    // filtered px
            o[1] = c[1];    // filtered py
            o[2] = mz;      // z passthrough
        }
    }
}

extern "C" void kernel_launch(void* const* d_in, const int* in_sizes, int n_in,
                              void* d_out, int out_size, void* d_ws, size_t ws_size,
                              hipStream_t stream) {
    const float* pos = (const float*)d_in[0];
    float* out  = (float*)d_out;
    float* ktab = (float*)d_ws;          // 64 KB gain table

    ekf_gains<<<1, 32, 0, stream>>>(ktab);
    ekf_filter<<<NBATCH / 64, 128, 0, stream>>>(pos, ktab, out);
}